// SSMSequenceMixer_37366215475478
// MI455X (gfx1250) — compile-verified
//
#include <hip/hip_runtime.h>
#include <math.h>
#include <stdint.h>

// ---------------------------------------------------------------------------
// SSM sequence mixer for MI455X (gfx1250, wave32, WMMA).
//
//   u      = x @ W_in^T + b_in           GEMM  M=8192 N=4096 K=2048 (f32)
//   states = scan_s tanh(u + st*gamma + beta)   (elementwise, serial in S)
//   y      = states @ W_out^T + b_out    GEMM  M=8192 N=2048 K=4096 (f32)
//
// Both GEMMs are "NT" layout (K contiguous in A and B). f32 inputs -> exact
// V_WMMA_F32_16X16X4_F32 path (matches reference numerics; ~275 GFLOP total,
// well under the f32 matrix-core roof, so data movement dominates).
//
// Data movement: LDS tiles are DOUBLE-BUFFERED and staged with
// GLOBAL_LOAD_ASYNC_TO_LDS_B128 (ASYNCcnt-tracked), overlapping next-stage
// global loads with the current stage's 64 WMMAs; one s_wait_asynccnt +
// barrier per stage. Fallback (builtin absent): register double-buffering.
// ---------------------------------------------------------------------------

typedef __attribute__((ext_vector_type(2))) float v2f;
typedef __attribute__((ext_vector_type(8))) float v8f;
typedef int v4i_g __attribute__((vector_size(16)));   // matches builtin param

#define DIM_    2048
#define HIDDEN_ 4096
#define BATCH_  4
#define SEQ_    2048
#define MTOT_   (BATCH_ * SEQ_)

#define BM 128
#define BN 128
#define BK 32
#define LDSP (BK + 4)   // stride 36 floats = 144B: 16B aligned, 64-bank conflict free

#if __has_builtin(__builtin_amdgcn_global_load_async_to_lds_b128) && \
    __has_builtin(__builtin_amdgcn_s_wait_asynccnt)
#define USE_ASYNC_LDS 1
#else
#define USE_ASYNC_LDS 0
#endif

#if USE_ASYNC_LDS
// Prototype (from clang diagnostics):
//   void __builtin_amdgcn_global_load_async_to_lds_b128(
//       v4i_g* gsrc /*generic*/, __shared__ v4i_g* ldst, int offset, int cpol)
#define ASYNC_COPY16(gsrc, ldst)                                                \
    __builtin_amdgcn_global_load_async_to_lds_b128(                             \
        (v4i_g*)(uintptr_t)(gsrc),                                              \
        (__attribute__((address_space(3))) v4i_g*)(uint32_t)(uintptr_t)(ldst),  \
        0, 0)
#endif

// C[M,N] = A[M,K] * B[N,K]^T + bias[N]   (all row-major, f32)
__global__ __launch_bounds__(256)
void gemm_nt_bias_f32(const float* __restrict__ A, const float* __restrict__ Bm,
                      const float* __restrict__ bias, float* __restrict__ C,
                      int M, int N, int K)
{
    __shared__ __align__(16) float As[2][BM][LDSP];
    __shared__ __align__(16) float Bs[2][BN][LDSP];

    const int tid  = threadIdx.x;
    const int lane = tid & 31;
    const int wave = tid >> 5;        // 8 waves: 4 along M x 2 along N
    const int wm   = wave & 3;        // M offset = wm*32
    const int wn   = wave >> 2;       // N offset = wn*64
    const int m0   = blockIdx.y * BM;
    const int n0   = blockIdx.x * BN;

    v8f acc[2][4] = {};               // 8 x 16x16 f32 accumulators (64 VGPRs)

    // A-fragment K sub-lane offset: lanes 0-15 hold K+0,K+1; lanes 16-31 K+2,K+3
    const int klo = (lane >> 4) << 1;
    const int l15 = lane & 15;

    // Per-thread staging slots: 1024 float4 slots per 128x32 tile, 4 per thread
    const int srow[4] = { (tid + 0)   >> 3, (tid + 256) >> 3,
                          (tid + 512) >> 3, (tid + 768) >> 3 };
    const int scol    = (tid & 7) << 2;

#if USE_ASYNC_LDS
    // ---- prologue: async-stage K-block 0 into buffer 0 ----
    #pragma unroll
    for (int it = 0; it < 4; ++it) {
        ASYNC_COPY16(&A[(size_t)(m0 + srow[it]) * K + scol], &As[0][srow[it]][scol]);
        ASYNC_COPY16(&Bm[(size_t)(n0 + srow[it]) * K + scol], &Bs[0][srow[it]][scol]);
    }
    __builtin_amdgcn_s_wait_asynccnt(0);
    __syncthreads();
#else
    float4 ra[4], rb[4];
    #pragma unroll
    for (int it = 0; it < 4; ++it) {
        ra[it] = *(const float4*)&A[(size_t)(m0 + srow[it]) * K + scol];
        rb[it] = *(const float4*)&Bm[(size_t)(n0 + srow[it]) * K + scol];
    }
    #pragma unroll
    for (int it = 0; it < 4; ++it) {
        *(float4*)&As[0][srow[it]][scol] = ra[it];
        *(float4*)&Bs[0][srow[it]][scol] = rb[it];
    }
    __syncthreads();
#endif

    int cur = 0;
    for (int k0 = 0; k0 < K; k0 += BK) {
        const bool more = (k0 + BK) < K;

#if USE_ASYNC_LDS
        // ---- issue async staging of next K-block into the other buffer ----
        if (more) {
            #pragma unroll
            for (int it = 0; it < 4; ++it) {
                ASYNC_COPY16(&A[(size_t)(m0 + srow[it]) * K + k0 + BK + scol],
                             &As[cur ^ 1][srow[it]][scol]);
                ASYNC_COPY16(&Bm[(size_t)(n0 + srow[it]) * K + k0 + BK + scol],
                             &Bs[cur ^ 1][srow[it]][scol]);
            }
        }
#else
        if (more) {
            #pragma unroll
            for (int it = 0; it < 4; ++it) {
                ra[it] = *(const float4*)&A[(size_t)(m0 + srow[it]) * K + k0 + BK + scol];
                rb[it] = *(const float4*)&Bm[(size_t)(n0 + srow[it]) * K + k0 + BK + scol];
            }
        }
#endif

        // ---- 8 K-steps of 4, 8 WMMAs each: v_wmma_f32_16x16x4_f32 ----
        #pragma unroll
        for (int kk = 0; kk < BK; kk += 4) {
            const int kl = kk + klo;
            v2f a[2], b[4];
            #pragma unroll
            for (int i = 0; i < 2; ++i)
                a[i] = *(const v2f*)&As[cur][wm * 32 + i * 16 + l15][kl];
            #pragma unroll
            for (int j = 0; j < 4; ++j)
                b[j] = *(const v2f*)&Bs[cur][wn * 64 + j * 16 + l15][kl];
            #pragma unroll
            for (int i = 0; i < 2; ++i)
                #pragma unroll
                for (int j = 0; j < 4; ++j)
                    acc[i][j] = __builtin_amdgcn_wmma_f32_16x16x4_f32(
                        /*neg_a=*/false, a[i], /*neg_b=*/false, b[j],
                        /*c_mod=*/(short)0, acc[i][j],
                        /*reuse_a=*/false, /*reuse_b=*/false);
        }

        if (more) {
#if USE_ASYNC_LDS
            __builtin_amdgcn_s_wait_asynccnt(0);   // next buffer fully landed
#else
            #pragma unroll
            for (int it = 0; it < 4; ++it) {
                *(float4*)&As[cur ^ 1][srow[it]][scol] = ra[it];
                *(float4*)&Bs[cur ^ 1][srow[it]][scol] = rb[it];
            }
#endif
            __syncthreads();   // also orders this stage's reads vs next writes
        }
        cur ^= 1;
    }

    // ---- epilogue: C/D layout = VGPR r -> M=r (lanes 0-15) / M=r+8 (16-31) ----
    #pragma unroll
    for (int i = 0; i < 2; ++i) {
        #pragma unroll
        for (int j = 0; j < 4; ++j) {
            const int rowbase = m0 + wm * 32 + i * 16 + ((lane >= 16) ? 8 : 0);
            const int col     = n0 + wn * 64 + j * 16 + l15;
            const float bv    = bias[col];
            #pragma unroll
            for (int r = 0; r < 8; ++r)
                C[(size_t)(rowbase + r) * N + col] = acc[i][j][r] + bv;
        }
    }
}

// In-place sequential recurrence over S: one thread per (b,h).
// U holds u on entry, states on exit.
__global__ __launch_bounds__(256)
void ssm_recurrence(float* __restrict__ U,
                    const float* __restrict__ gamma,
                    const float* __restrict__ beta)
{
    const int idx = blockIdx.x * blockDim.x + threadIdx.x;   // 0 .. B*H-1
    const int b = idx / HIDDEN_;
    const int h = idx - b * HIDDEN_;
    const float g  = gamma[h];
    const float bt = beta[h];
    float st = 0.0f;
    float* p = U + (size_t)b * SEQ_ * HIDDEN_ + h;
    for (int s = 0; s < SEQ_; ++s) {
        float u = *p;
        st = tanhf(fmaf(st, g, bt) + u);
        *p = st;
        p += HIDDEN_;
    }
}

extern "C" void kernel_launch(void* const* d_in, const int* in_sizes, int n_in,
                              void* d_out, int out_size, void* d_ws, size_t ws_size,
                              hipStream_t stream)
{
    const float* x     = (const float*)d_in[0];   // [B,S,D]
    const float* W_in  = (const float*)d_in[1];   // [H,D]
    const float* b_in  = (const float*)d_in[2];   // [H]
    const float* W_out = (const float*)d_in[3];   // [D,H]
    const float* b_out = (const float*)d_in[4];   // [D]
    const float* gamma = (const float*)d_in[5];   // [H]
    const float* beta  = (const float*)d_in[6];   // [H]
    float*       y     = (float*)d_out;           // [B,S,D]
    float*       U     = (float*)d_ws;            // [B,S,H] = 128 MB scratch

    // GEMM 1: U = x @ W_in^T + b_in   (M=8192, N=4096, K=2048)
    {
        dim3 grid(HIDDEN_ / BN, MTOT_ / BM);
        gemm_nt_bias_f32<<<grid, 256, 0, stream>>>(x, W_in, b_in, U,
                                                   MTOT_, HIDDEN_, DIM_);
    }

    // Recurrence (in place over U): 16384 threads
    {
        dim3 grid((BATCH_ * HIDDEN_) / 256);
        ssm_recurrence<<<grid, 256, 0, stream>>>(U, gamma, beta);
    }

    // GEMM 2: y = states @ W_out^T + b_out   (M=8192, N=2048, K=4096)
    {
        dim3 grid(DIM_ / BN, MTOT_ / BM);
        gemm_nt_bias_f32<<<grid, 256, 0, stream>>>(U, W_out, b_out, y,
                                                   MTOT_, DIM_, HIDDEN_);
    }
}